// NeuralLDPCDecoder_52407190945949
// MI455X (gfx1250) — compile-verified
//
#include <hip/hip_runtime.h>
#include <hip/hip_bf16.h>
#include <stdint.h>

// QC-LDPC neural min-sum decoder, structured implementation for MI455X.
// N=32 vars, M=16 checks, Z=32 lifting, DV=5, DC=10, E=160, B=32, 5 iters.
#define NN   32
#define MM   16
#define ZL   32
#define DV_  5
#define DC_  10
#define EE   160
#define EES  162            // padded LDS row stride: 162 mod 64 = 34 -> conflict-free
#define BB   32
#define IT   5
#define EZ   (EE * ZL)
#define YS   (ZL + 1)       // y tile stride 33, coprime with 64 banks
#define NEG_FILL 10000.0f

typedef __attribute__((ext_vector_type(2))) float v2f;
typedef __attribute__((ext_vector_type(8))) float v8f;

// ---------------------------------------------------------------------------
// Setup: extract graph structure from the dense masks / permutation matrices.
// Reads only 160*32 elements of L1 (row e*Z holds its 1 at col e*Z+shift[e]),
// never the 105MB matrices in full.
// ws layout (ints): shift[E] | var_of[E] | var_edges[N*DV] | partners[E*(DC-1)]
// ---------------------------------------------------------------------------
__global__ void ldpc_setup(const float* __restrict__ Wskip,     // [N,E]
                           const float* __restrict__ We2o,      // [E,E]
                           const float* __restrict__ L1,        // [EZ,EZ]
                           int* __restrict__ ws) {
  const int t = threadIdx.x;
  int* shift     = ws;
  int* var_of    = ws + EE;
  int* var_edges = ws + 2 * EE;             // [N][DV]
  int* partners  = ws + 2 * EE + NN * DV_;  // [E][DC-1]

  if (t < EE) {
    // shift[e]: scan Z columns of L1 row e*Z within its own edge block.
    const float* row = L1 + (size_t)(t * ZL) * EZ + (size_t)t * ZL;
    int s = 0;
    for (int c = 0; c < ZL; ++c)
      if (row[c] > 0.5f) s = c;
    shift[t] = s;

    // var_of[e]: the single row of W_skip with a 1 in column e.
    int v = 0;
    for (int r = 0; r < NN; ++r)
      if (Wskip[r * EE + t] > 0.5f) v = r;
    var_of[t] = v;

    // partners of edge i: j with Wt[i,j] = We2o[j,i] == 1 (same check, j != i).
    int cnt = 0;
    for (int j = 0; j < EE; ++j) {
      if (j != t && We2o[j * EE + t] > 0.5f) {
        if (cnt < DC_ - 1) partners[t * (DC_ - 1) + cnt] = j;
        ++cnt;
      }
    }
    for (; cnt < DC_ - 1; ++cnt) partners[t * (DC_ - 1) + cnt] = t;
  }
  if (t < NN) {
    int cnt = 0;
    for (int e = 0; e < EE; ++e) {
      if (Wskip[t * EE + e] > 0.5f) {
        if (cnt < DV_) var_edges[t * DV_ + cnt] = e;
        ++cnt;
      }
    }
    for (; cnt < DV_; ++cnt) var_edges[t * DV_ + cnt] = 0;
  }
}

// ---------------------------------------------------------------------------
// Main decoder: one workgroup (8 wave32) per batch element, LDS-resident.
// ---------------------------------------------------------------------------
__global__ __launch_bounds__(256)
void ldpc_decode(const float* __restrict__ xa,    // [B,N,Z]
                 const float* __restrict__ Wout,  // [E,N]
                 const float* __restrict__ wvar,  // [IT,E]
                 const float* __restrict__ bvar,  // [IT,E]
                 const int*   __restrict__ ws,
                 float* __restrict__ out) {       // [IT,B,N*Z]
  __shared__ __align__(16) float xa_s[NN * ZL];   // [n][z]
  __shared__ float tv[ZL * NN];                   // [z][v]  var-node totals
  __shared__ float y_s[NN * YS];                  // [v][z]  staged marginal tile
  __shared__ float llr[ZL * EES];                 // [z][e]  (padded rows)
  __shared__ float x2s[ZL * EES];                 // [z][e]  post-perm1 messages
  __shared__ float xo_s[ZL * EES];                // [z][e]  pre-perm2 check outputs
  __shared__ float woutT[NN * EES];               // [n][e]  W_output transposed
  __shared__ int   shift_s[EE];
  __shared__ int   varof_s[EE];
  __shared__ int   vedge_s[NN * DV_];
  __shared__ int   part_s[EE * (DC_ - 1)];

  const int b    = blockIdx.x;
  const int t    = threadIdx.x;
  const int wave = t >> 5;
  const int lane = t & 31;

  // ---- async copy of this block's channel LLRs: one B128 per thread,
  //      global -> LDS without touching VGPRs (ASYNCcnt-tracked).
  {
    const float* src = xa + (size_t)b * NN * ZL;          // uniform -> SGPR pair
    unsigned lds_off = (unsigned)(uintptr_t)(&xa_s[t * 4]); // low 32b = LDS offset
    unsigned goff    = (unsigned)(t * 16);                  // per-lane byte offset
    asm volatile("global_load_async_to_lds_b128 %0, %1, %2 offset:0"
                 :: "v"(lds_off), "v"(goff), "s"(src) : "memory");
  }

  // ---- regular loads (transposing W_output into [n][e] for vectorized B frags)
  for (int i = t; i < EE * NN; i += 256) {
    const int e = i >> 5, n = i & 31;
    woutT[n * EES + e] = Wout[i];
  }
  for (int i = t; i < ZL * EES; i += 256) llr[i] = 0.0f;
  for (int i = t; i < EE; i += 256) { shift_s[i] = ws[i]; varof_s[i] = ws[EE + i]; }
  for (int i = t; i < NN * DV_; i += 256) vedge_s[i] = ws[2 * EE + i];
  for (int i = t; i < EE * (DC_ - 1); i += 256) part_s[i] = ws[2 * EE + NN * DV_ + i];

  asm volatile("s_wait_asynccnt 0" ::: "memory");
  __syncthreads();

  for (int it = 0; it < IT; ++it) {
    // ---- var-node totals: tv[z][v] = xa_in[z][v] + sum_{e in edges(v)} llr[z][e]
    for (int i = t; i < ZL * NN; i += 256) {
      const int z = i >> 5, v = i & 31;
      float s = xa_s[v * ZL + z];
#pragma unroll
      for (int k = 0; k < DV_; ++k) s += llr[z * EES + vedge_s[v * DV_ + k]];
      tv[i] = s;
    }
    __syncthreads();

    // ---- extrinsic var->check + lifting perm1 (read at (z+shift)%Z)
    for (int i = t; i < ZL * EE; i += 256) {
      const int z = i / EE, e = i - z * EE;
      const int zp = (z + shift_s[e]) & (ZL - 1);
      x2s[z * EES + e] = tv[zp * NN + varof_s[e]] - llr[zp * EES + e];
    }
    __syncthreads();

    // ---- check-node min-sum over the 9 same-check partners
    for (int i = t; i < ZL * EE; i += 256) {
      const int z = i / EE, e = i - z * EE;
      float mn = NEG_FILL;
      int pos = 0;
#pragma unroll
      for (int k = 0; k < DC_ - 1; ++k) {
        const float v = x2s[z * EES + part_s[e * (DC_ - 1) + k]];
        const float a = fabsf(v);
        mn = fminf(mn, (a > 0.0f) ? a : NEG_FILL);
        pos += (v > 0.0f) ? 1 : 0;
      }
      // sign(-prod(1-2*(x>0))) == +1 iff #positives odd
      xo_s[z * EES + e] = mn * ((pos & 1) ? 1.0f : -1.0f);
    }
    __syncthreads();

    // ---- lifting perm2 + learned scale/offset with ReLU on magnitude
    for (int i = t; i < ZL * EE; i += 256) {
      const int z = i / EE, e = i - z * EE;
      const int zp = (z + shift_s[e]) & (ZL - 1);
      const float v = xo_s[zp * EES + e];
      const float mag = fmaxf(fabsf(v) * wvar[it * EE + e] + bvar[it * EE + e], 0.0f);
      const float sg = (v > 0.0f) ? 1.0f : ((v < 0.0f) ? -1.0f : 0.0f);
      llr[z * EES + e] = mag * sg;
    }
    __syncthreads();

    // ---- marginal: y[z][v] = llr[z][:] @ Wout[:][v] via V_WMMA_F32_16X16X4_F32.
    // 2x2 tiles of 16x16 on waves 0..3 (wave-uniform branch keeps EXEC all-1s);
    // K = 160 in steps of 4; all fragment loads are b64 and bank-conflict-free.
    if (wave < 4) {
      const int zt   = (wave >> 1) * 16;  // M (= z) tile base
      const int vt   = (wave & 1) * 16;   // N (= v) tile base
      const int half = lane >> 4;         // lane half selects K pair / row group
      const int l16  = lane & 15;
      const float* arow = &llr[(zt + l16) * EES];        // A row for this lane
      const float* brow = &woutT[(vt + l16) * EES];      // B column (transposed)
      v8f acc = {};
      for (int k4 = 0; k4 < EE; k4 += 4) {
        // A 16x4: VGPR0 = {K0 | K2}, VGPR1 = {K1 | K3}; M = lane%16
        const v2f av = *(const v2f*)(arow + k4 + 2 * half);
        // B 4x16: VGPR0 = rows {K0 | K2}, VGPR1 = rows {K1 | K3}; N = lane%16
        const v2f bv = *(const v2f*)(brow + k4 + 2 * half);
        acc = __builtin_amdgcn_wmma_f32_16x16x4_f32(false, av, false, bv,
                                                    (short)0, acc, false, false);
      }
      // D layout: VGPR r holds row M = r (lanes 0-15) / r+8 (lanes 16-31).
      // Stage into bank-padded LDS so the global store below is coalesced.
#pragma unroll
      for (int r = 0; r < 8; ++r)
        y_s[(vt + l16) * YS + zt + r + 8 * half] = acc[r];
    }
    __syncthreads();

    // ---- coalesced output: out[it][b][n*Z+z] = xa + y
    float* outp = out + (size_t)it * BB * NN * ZL + (size_t)b * NN * ZL;
    for (int i = t; i < NN * ZL; i += 256) {
      const int v = i >> 5, z = i & 31;
      outp[i] = xa_s[i] + y_s[v * YS + z];
    }
    __syncthreads();
  }
}

// ---------------------------------------------------------------------------
extern "C" void kernel_launch(void* const* d_in, const int* in_sizes, int n_in,
                              void* d_out, int out_size, void* d_ws, size_t ws_size,
                              hipStream_t stream) {
  const float* xa    = (const float*)d_in[0];  // [B,N,Z]
  const float* Wskip = (const float*)d_in[1];  // [N,E]
  // d_in[2] = W_odd2even  (structure recovered from W_skip: same-var adjacency)
  const float* We2o  = (const float*)d_in[3];  // [E,E]
  const float* Wout  = (const float*)d_in[4];  // [E,N]
  const float* L1    = (const float*)d_in[5];  // [EZ,EZ] (only 160x32 elems read)
  // d_in[6] = L2 (its action equals reading (z+shift)%Z; derived from L1)
  const float* wv    = (const float*)d_in[7];  // [IT,E]
  const float* bv    = (const float*)d_in[8];  // [IT,E]
  int* ws = (int*)d_ws;

  ldpc_setup<<<1, 256, 0, stream>>>(Wskip, We2o, L1, ws);
  ldpc_decode<<<BB, 256, 0, stream>>>(xa, Wout, wv, bv, ws, (float*)d_out);
}